// GeoGCNLayer_48893907697759
// MI455X (gfx1250) — compile-verified
//
#include <hip/hip_runtime.h>

typedef __attribute__((ext_vector_type(2))) float v2f;
typedef __attribute__((ext_vector_type(8))) float v8f;

#define D   128
#define H4D 512
#define MB  5      // row-tiles (of 16 rows) per wave: 5x B-fragment reuse

__device__ __forceinline__ void atomic_add_f32(float* p, float v) {
    __hip_atomic_fetch_add(p, v, __ATOMIC_RELAXED, __HIP_MEMORY_SCOPE_AGENT);
}

// ---------------------------------------------------------------------------
// Kernel 1: f = feat @ W_tran^T  (fp32 WMMA 16x16x4, one wave = MB*16 rows)
// A-frag (16x4 f32): lanes 0-15 hold K=k0,k0+1 ; lanes 16-31 hold K=k0+2,k0+3
// B-frag (4x16): lane n holds B[k,n] = W[n0+n, k] (same per-lane k split)
// C/D: vgpr v -> row m0+v (lanes 0-15) / m0+v+8 (lanes 16-31), col n0+(lane&15)
// Each B fragment is reused by MB WMMAs -> 5x less W traffic.
// ---------------------------------------------------------------------------
__global__ void __launch_bounds__(256)
transform_wmma_kernel(const float* __restrict__ feat,
                      const float* __restrict__ W,   // [D,D] row-major W[n,k]
                      float* __restrict__ f,
                      int n_tiles)                    // N/16
{
    const int lane  = threadIdx.x & 31;
    const int wave  = threadIdx.x >> 5;
    const int nchnk = (n_tiles + MB - 1) / MB;
    const int chunk = blockIdx.x * (blockDim.x >> 5) + wave;
    if (chunk >= nchnk) return;                       // wave-uniform

    const int half = lane >> 4;
    const int l16  = lane & 15;

    const float* arow[MB];
    bool valid[MB];
#pragma unroll
    for (int m = 0; m < MB; ++m) {
        const int t = chunk * MB + m;
        valid[m] = (t < n_tiles);                     // wave-uniform
        const int tu = valid[m] ? t : 0;
        arow[m] = feat + (size_t)(tu * 16 + l16) * D;
    }

    for (int nt = 0; nt < D / 16; ++nt) {
        const int n0 = nt * 16;
        const float* brow = W + (size_t)(n0 + l16) * D;   // W[n, :]
        v8f c[MB];
#pragma unroll
        for (int m = 0; m < MB; ++m) c[m] = (v8f){};
#pragma unroll
        for (int k0 = 0; k0 < D; k0 += 4) {
            const int ka = k0 + half * 2;
            v2f b;
            b.x = brow[ka];  b.y = brow[ka + 1];
#pragma unroll
            for (int m = 0; m < MB; ++m) {
                v2f a;
                a.x = arow[m][ka];  a.y = arow[m][ka + 1];
                c[m] = __builtin_amdgcn_wmma_f32_16x16x4_f32(
                           false, a, false, b, (short)0, c[m], false, false);
            }
        }
#pragma unroll
        for (int m = 0; m < MB; ++m) {
            if (valid[m]) {                           // wave-uniform
                const int m0 = (chunk * MB + m) * 16;
#pragma unroll
                for (int v = 0; v < 8; ++v)
                    f[(size_t)(m0 + v + half * 8) * D + n0 + l16] = c[m][v];
            }
        }
    }
}

// ---------------------------------------------------------------------------
// Kernel 2: edge scatter.  One wave per edge: 32 lanes x float4 = 128 floats.
//   sum[dst] += f[src] * w   (w==null -> 1.0)     cnt[dst] += 1  (if cnt)
// Destinations (3 x 51 MB) are L2-resident -> hw global_atomic_add_f32 path.
// ---------------------------------------------------------------------------
__global__ void __launch_bounds__(256)
edge_scatter_kernel(const float* __restrict__ f,
                    const int* __restrict__ src,
                    const int* __restrict__ dst,
                    const float* __restrict__ w,      // may be null
                    float* __restrict__ sum,
                    float* __restrict__ cnt,          // may be null
                    int n_edges)
{
    const int lane = threadIdx.x & 31;
    const int e = blockIdx.x * (blockDim.x >> 5) + (threadIdx.x >> 5);
    if (e >= n_edges) return;                         // wave-uniform

    const int s = src[e];
    const int d = dst[e];
    const float wt = w ? w[e] : 1.0f;

    const float4 v = *reinterpret_cast<const float4*>(f + (size_t)s * D + lane * 4);
    float* o = sum + (size_t)d * D + lane * 4;
    atomic_add_f32(o + 0, v.x * wt);
    atomic_add_f32(o + 1, v.y * wt);
    atomic_add_f32(o + 2, v.z * wt);
    atomic_add_f32(o + 3, v.w * wt);
    if (cnt != nullptr && lane == 0) atomic_add_f32(cnt + d, 1.0f);
}

// ---------------------------------------------------------------------------
// Kernel 3: sum /= max(cnt,1)
// ---------------------------------------------------------------------------
__global__ void __launch_bounds__(256)
normalize_kernel(float* __restrict__ sum, const float* __restrict__ cnt, size_t nd)
{
    const size_t i = (size_t)blockIdx.x * blockDim.x + threadIdx.x;
    if (i >= nd) return;
    const float c = cnt[i >> 7];
    sum[i] = sum[i] / fmaxf(c, 1.0f);
}

// ---------------------------------------------------------------------------
// Kernel 4: fused semantic attention reduction.
//   H = tanh(Z @ W1^T + b1) via fp32 WMMA (MB row-tiles per wave),
//   contracted on the fly with w2 and mean-reduced into s[r] via atomics.
//   H (614 MB) is never materialized; each W1 fragment feeds MB WMMAs.
// ---------------------------------------------------------------------------
__global__ void __launch_bounds__(256)
attention_reduce_kernel(const float* __restrict__ geo,
                        const float* __restrict__ trans,
                        const float* __restrict__ cat,
                        const float* __restrict__ w1,  // [512,128]
                        const float* __restrict__ b1,  // [512]
                        const float* __restrict__ w2,  // [512]
                        float* __restrict__ s_accum,   // [3]
                        int n_tiles)                   // N/16
{
    const int lane  = threadIdx.x & 31;
    const int wave  = threadIdx.x >> 5;
    const int nchnk = (n_tiles + MB - 1) / MB;
    const int gt = blockIdx.x * (blockDim.x >> 5) + wave;
    if (gt >= 3 * nchnk) return;                      // wave-uniform

    const int r  = gt / nchnk;
    const int tc = gt % nchnk;
    const float* z = (r == 0) ? geo : ((r == 1) ? trans : cat);

    const int half = lane >> 4;
    const int l16  = lane & 15;

    const float* arow[MB];
    bool valid[MB];
#pragma unroll
    for (int m = 0; m < MB; ++m) {
        const int t = tc * MB + m;
        valid[m] = (t < n_tiles);                     // wave-uniform
        const int tu = valid[m] ? t : 0;
        arow[m] = z + (size_t)(tu * 16 + l16) * D;
    }

    float acc = 0.0f;
    for (int ht = 0; ht < H4D / 16; ++ht) {
        const int h0 = ht * 16;
        const float* brow = w1 + (size_t)(h0 + l16) * D;  // att_w1[h, :]
        v8f c[MB];
#pragma unroll
        for (int m = 0; m < MB; ++m) c[m] = (v8f){};
#pragma unroll
        for (int k0 = 0; k0 < D; k0 += 4) {
            const int ka = k0 + half * 2;
            v2f b;
            b.x = brow[ka];  b.y = brow[ka + 1];
#pragma unroll
            for (int m = 0; m < MB; ++m) {
                v2f a;
                a.x = arow[m][ka];  a.y = arow[m][ka + 1];
                c[m] = __builtin_amdgcn_wmma_f32_16x16x4_f32(
                           false, a, false, b, (short)0, c[m], false, false);
            }
        }
        const int col = h0 + l16;
        const float bb = b1[col];
        const float ww = w2[col];
#pragma unroll
        for (int m = 0; m < MB; ++m) {
            if (valid[m]) {                           // wave-uniform
#pragma unroll
                for (int v = 0; v < 8; ++v)
                    acc += tanhf(c[m][v] + bb) * ww;  // v_tanh_f32 TRANS ops
            }
        }
    }
    // wave32 reduction
#pragma unroll
    for (int off = 16; off > 0; off >>= 1)
        acc += __shfl_xor(acc, off, 32);
    if (lane == 0) atomic_add_f32(&s_accum[r], acc);
}

// ---------------------------------------------------------------------------
// Kernel 5: beta = softmax(s / N)   (single thread; trivially cheap)
// ---------------------------------------------------------------------------
__global__ void softmax3_kernel(const float* __restrict__ s, float* __restrict__ beta,
                                float inv_n)
{
    if (threadIdx.x == 0 && blockIdx.x == 0) {
        float w0 = s[0] * inv_n, w1 = s[1] * inv_n, w2 = s[2] * inv_n;
        float m = fmaxf(w0, fmaxf(w1, w2));
        float e0 = expf(w0 - m), e1 = expf(w1 - m), e2 = expf(w2 - m);
        float inv = 1.0f / (e0 + e1 + e2);
        beta[0] = e0 * inv; beta[1] = e1 * inv; beta[2] = e2 * inv;
    }
}

// ---------------------------------------------------------------------------
// Kernel 6: out = leaky_relu(b0*geo + b1*trans + b2*cat, 0.2)
// ---------------------------------------------------------------------------
__global__ void __launch_bounds__(256)
fuse_kernel(const float* __restrict__ geo,
            const float* __restrict__ trans,
            const float* __restrict__ cat,
            const float* __restrict__ beta,
            float* __restrict__ out, size_t nd)
{
    const size_t i = (size_t)blockIdx.x * blockDim.x + threadIdx.x;
    if (i >= nd) return;
    const float x = beta[0] * geo[i] + beta[1] * trans[i] + beta[2] * cat[i];
    out[i] = (x >= 0.0f) ? x : 0.2f * x;
}

// ---------------------------------------------------------------------------
extern "C" void kernel_launch(void* const* d_in, const int* in_sizes, int n_in,
                              void* d_out, int out_size, void* d_ws, size_t ws_size,
                              hipStream_t stream) {
    const float* feat   = (const float*)d_in[0];
    const float* W_tran = (const float*)d_in[1];
    const float* att_w1 = (const float*)d_in[2];
    const float* att_b1 = (const float*)d_in[3];
    const float* att_w2 = (const float*)d_in[4];
    const float* tw     = (const float*)d_in[5];
    const int* geo_src  = (const int*)d_in[6];
    const int* geo_dst  = (const int*)d_in[7];
    const int* cat_src  = (const int*)d_in[8];
    const int* cat_dst  = (const int*)d_in[9];
    const int* trn_src  = (const int*)d_in[10];
    const int* trn_dst  = (const int*)d_in[11];
    float* out = (float*)d_out;

    const int N = in_sizes[0] / D;     // 100000
    const int E = in_sizes[6];         // 600000
    const size_t ND = (size_t)N * D;

    // Workspace layout (floats)
    float* ws    = (float*)d_ws;
    float* f     = ws;                 // [N,D]
    float* geo   = ws + ND;            // [N,D]
    float* trans = ws + 2 * ND;        // [N,D]
    float* cat   = ws + 3 * ND;        // [N,D]
    float* cntg  = ws + 4 * ND;        // [N]
    float* cntc  = cntg + N;           // [N]
    float* sacc  = cntc + N;           // [3]
    float* beta  = sacc + 4;           // [3]

    // Zero accumulators (geo..beta contiguous) — capture-safe async memset.
    hipMemsetAsync(geo, 0, (3 * ND + 2 * (size_t)N + 8) * sizeof(float), stream);

    const int n_tiles = N / 16;                           // 6250
    const int nchunks = (n_tiles + MB - 1) / MB;          // 1250
    const int wpb = 8;                                    // waves per 256-thr block

    // 1) node transform (WMMA fp32, MB-blocked)
    transform_wmma_kernel<<<(nchunks + wpb - 1) / wpb, 256, 0, stream>>>(
        feat, W_tran, f, n_tiles);

    // 2) three edge aggregations
    const int eblocks = (E + wpb - 1) / wpb;
    edge_scatter_kernel<<<eblocks, 256, 0, stream>>>(f, geo_src, geo_dst, nullptr, geo,  cntg, E);
    edge_scatter_kernel<<<eblocks, 256, 0, stream>>>(f, cat_src, cat_dst, nullptr, cat,  cntc, E);
    edge_scatter_kernel<<<eblocks, 256, 0, stream>>>(f, trn_src, trn_dst, tw,      trans, nullptr, E);

    // 3) mean normalization for geo & cat
    const int nblocks = (int)((ND + 255) / 256);
    normalize_kernel<<<nblocks, 256, 0, stream>>>(geo, cntg, ND);
    normalize_kernel<<<nblocks, 256, 0, stream>>>(cat, cntc, ND);

    // 4) fused attention GEMM + tanh + dot + mean-reduce (WMMA fp32, MB-blocked)
    attention_reduce_kernel<<<(3 * nchunks + wpb - 1) / wpb, 256, 0, stream>>>(
        geo, trans, cat, att_w1, att_b1, att_w2, sacc, n_tiles);

    // 5) softmax over the 3 relation scores
    softmax3_kernel<<<1, 32, 0, stream>>>(sacc, beta, 1.0f / (float)N);

    // 6) fuse + leaky_relu
    fuse_kernel<<<nblocks, 256, 0, stream>>>(geo, trans, cat, beta, out, ND);
}